// GATLayer_9577777070577
// MI455X (gfx1250) — compile-verified
//
#include <hip/hip_runtime.h>
#include <hip/hip_bf16.h>
#include <math.h>

// ---------------- problem constants (from reference) ----------------
#define IN_C   256
#define OUT_C  64
#define HEADS  4
#define HC     (HEADS * OUT_C)   // 256 = projected width
#define NEG_SLOPE 0.2f

typedef __attribute__((ext_vector_type(2))) float v2f;
typedef __attribute__((ext_vector_type(8))) float v8f;

// ===================================================================
// Kernel 1: xh = x @ W  via V_WMMA_F32_16X16X4_F32 (fp32 matrix pipe)
// One wave computes one 16x16 output tile; K-loop over 256 in steps
// of 4.  A (16x4 f32) layout per ISA: lanes 0-15 hold K=0/1 in
// VGPR0/1, lanes 16-31 hold K=2/3.  B (4x16) mirrors it; C/D uses the
// 8-VGPR row striping (VGPR r -> rows r and r+8).
// ===================================================================
__global__ void __launch_bounds__(256)
gat_gemm_wmma(const float* __restrict__ x, const float* __restrict__ W,
              float* __restrict__ xh, int n_rows) {
  const int wave   = blockIdx.x * (blockDim.x >> 5) + (threadIdx.x >> 5);
  const int lane   = threadIdx.x & 31;
  const int tilesN = HC / 16;                 // 16 column tiles
  const int mt = wave / tilesN;
  const int nt = wave % tilesN;
  const int m0 = mt * 16;
  if (m0 >= n_rows) return;
  const int n0   = nt * 16;
  const int half = lane >> 4;                 // 0: lanes 0-15, 1: lanes 16-31
  const int lr   = lane & 15;

  // clamp row for partial tiles (N=50000 is a multiple of 16, so this
  // is just safety; WMMA itself runs with full EXEC)
  const int arow = (m0 + lr < n_rows) ? (m0 + lr) : (n_rows - 1);
  const float* __restrict__ aptr = x + (size_t)arow * IN_C;

  v8f c = {};
  #pragma unroll 4
  for (int k = 0; k < IN_C; k += 4) {
    const int ka = k + half * 2;              // this lane's K pair
    v2f a, b;
    a.x = aptr[ka];
    a.y = aptr[ka + 1];
    b.x = W[(size_t)ka       * HC + n0 + lr];
    b.y = W[(size_t)(ka + 1) * HC + n0 + lr];
    // prefetch the next A K-group (lowers to global_prefetch_b8)
    if (k + 4 < IN_C) __builtin_prefetch(aptr + k + 4, 0, 3);
    c = __builtin_amdgcn_wmma_f32_16x16x4_f32(
        /*neg_a=*/false, a, /*neg_b=*/false, b,
        /*c_mod=*/(short)0, c, /*reuse_a=*/false, /*reuse_b=*/false);
  }

  // C/D layout: VGPR r holds row (r + 8*half), col lr
  const int rbase = m0 + half * 8;
  #pragma unroll
  for (int r = 0; r < 8; ++r) {
    const int row = rbase + r;
    if (row < n_rows) xh[(size_t)row * HC + n0 + lr] = c[r];
  }
}

// ===================================================================
// Kernel 2: per-(node, head) attention dots over 64 channels
// ===================================================================
__global__ void __launch_bounds__(256)
gat_att_dots(const float* __restrict__ xh,
             const float* __restrict__ att_src,
             const float* __restrict__ att_dst,
             float* __restrict__ a_src, float* __restrict__ a_dst, int n) {
  const int t = blockIdx.x * blockDim.x + threadIdx.x;
  if (t >= n * HEADS) return;
  const int node = t >> 2;
  const int h    = t & 3;
  const float* __restrict__ row = xh + (size_t)node * HC + h * OUT_C;
  const float* __restrict__ as  = att_src + h * OUT_C;
  const float* __restrict__ ad  = att_dst + h * OUT_C;
  float ss = 0.f, sd = 0.f;
  #pragma unroll 8
  for (int cix = 0; cix < OUT_C; ++cix) {
    const float v = row[cix];
    ss += v * as[cix];
    sd += v * ad[cix];
  }
  a_src[t] = ss;
  a_dst[t] = sd;
}

// ===================================================================
// Kernel 3: init accumulators (out=0, amax=-inf, denom=0)
// ===================================================================
__global__ void __launch_bounds__(256)
gat_init(float* __restrict__ out, float* __restrict__ amax,
         float* __restrict__ denom, int out_n, int nh) {
  const int t = blockIdx.x * blockDim.x + threadIdx.x;
  if (t < out_n) out[t] = 0.0f;
  if (t < nh) {
    amax[t]  = __uint_as_float(0xFF800000u);  // -inf
    denom[t] = 0.0f;
  }
}

// monotonic-bits float atomic max (global_atomic_max_i32 / min_u32)
__device__ __forceinline__ void atomicMaxFloat(float* addr, float val) {
  if (val >= 0.0f)
    atomicMax((int*)addr, __float_as_int(val));
  else
    atomicMin((unsigned int*)addr, (unsigned int)__float_as_int(val));
}

__device__ __forceinline__ void edge_endpoints(const int* __restrict__ ei,
                                               int e, int E, int& s, int& d) {
  if (e < E) { s = ei[e]; d = ei[E + e]; }
  else       { s = e - E; d = e - E; }      // appended self-loops
}

// ===================================================================
// Kernel 4: alpha = leakyrelu(a_src[s]+a_dst[d]); segment max (atomic)
// ===================================================================
__global__ void __launch_bounds__(256)
gat_edge_alpha_max(const int* __restrict__ ei,
                   const float* __restrict__ a_src,
                   const float* __restrict__ a_dst,
                   float* __restrict__ ealpha, float* __restrict__ amax,
                   int E, int EN) {
  const int e = blockIdx.x * blockDim.x + threadIdx.x;
  if (e >= EN) return;
  int s, d;
  edge_endpoints(ei, e, E, s, d);
  #pragma unroll
  for (int h = 0; h < HEADS; ++h) {
    float a = a_src[s * HEADS + h] + a_dst[d * HEADS + h];
    a = (a >= 0.0f) ? a : NEG_SLOPE * a;
    ealpha[(size_t)e * HEADS + h] = a;
    atomicMaxFloat(&amax[d * HEADS + h], a);
  }
}

// ===================================================================
// Kernel 5: ex = exp(alpha - amax[d]); segment sum (atomic)
// ===================================================================
__global__ void __launch_bounds__(256)
gat_edge_exp_sum(const int* __restrict__ ei,
                 float* __restrict__ ealpha,
                 const float* __restrict__ amax,
                 float* __restrict__ denom, int E, int EN) {
  const int e = blockIdx.x * blockDim.x + threadIdx.x;
  if (e >= EN) return;
  int s, d;
  edge_endpoints(ei, e, E, s, d);
  #pragma unroll
  for (int h = 0; h < HEADS; ++h) {
    const float ex = __expf(ealpha[(size_t)e * HEADS + h] - amax[d * HEADS + h]);
    ealpha[(size_t)e * HEADS + h] = ex;      // store numerator in-place
    atomicAdd(&denom[d * HEADS + h], ex);
  }
}

// ===================================================================
// Kernel 6: weighted scatter-add, head-mean folded in.
// 64 threads per edge (one per output channel):
//   out[d][c] += 0.25 * sum_h attn[h] * xh[s][h*64+c]
// -> only 64 f32 atomics per edge instead of 256; xh gathers are
//    contiguous across the 64 lanes and L2-resident (51 MB < 192 MB).
// ===================================================================
__global__ void __launch_bounds__(256)
gat_scatter(const int* __restrict__ ei,
            const float* __restrict__ xh,
            const float* __restrict__ ealpha,
            const float* __restrict__ denom,
            float* __restrict__ out, int E, int EN) {
  const long long t = (long long)blockIdx.x * blockDim.x + threadIdx.x;
  const int e = (int)(t >> 6);
  const int c = (int)(t & 63);
  if (e >= EN) return;
  int s, d;
  edge_endpoints(ei, e, E, s, d);
  const float* __restrict__ srow = xh + (size_t)s * HC;
  float acc = 0.0f;
  #pragma unroll
  for (int h = 0; h < HEADS; ++h) {
    const float attn = ealpha[(size_t)e * HEADS + h] / denom[d * HEADS + h];
    acc += attn * srow[h * OUT_C + c];
  }
  atomicAdd(&out[(size_t)d * OUT_C + c], 0.25f * acc);
}

// ===================================================================
// Kernel 7: out = relu(out + bias_gat + bias)
// ===================================================================
__global__ void __launch_bounds__(256)
gat_finalize(float* __restrict__ out, const float* __restrict__ bias_gat,
             const float* __restrict__ bias, int total) {
  const int t = blockIdx.x * blockDim.x + threadIdx.x;
  if (t >= total) return;
  const int c = t & (OUT_C - 1);
  const float v = out[t] + bias_gat[c] + bias[c];
  out[t] = fmaxf(v, 0.0f);
}

// ===================================================================
extern "C" void kernel_launch(void* const* d_in, const int* in_sizes, int n_in,
                              void* d_out, int out_size, void* d_ws, size_t ws_size,
                              hipStream_t stream) {
  const float* x        = (const float*)d_in[0];
  const int*   ei       = (const int*)  d_in[1];
  const float* W        = (const float*)d_in[2];
  const float* att_src  = (const float*)d_in[3];
  const float* att_dst  = (const float*)d_in[4];
  const float* bias_gat = (const float*)d_in[5];
  const float* bias     = (const float*)d_in[6];
  float*       out      = (float*)d_out;

  const int N  = in_sizes[0] / IN_C;   // 50000
  const int E  = in_sizes[1] / 2;      // 800000
  const int EN = E + N;                // edges + self-loops

  // workspace layout (floats): xh | a_src | a_dst | amax | denom | ealpha
  float* ws = (float*)d_ws;
  size_t off = 0;
  float* xh     = ws + off; off += (size_t)N * HC;
  float* a_src  = ws + off; off += (size_t)N * HEADS;
  float* a_dst  = ws + off; off += (size_t)N * HEADS;
  float* amax   = ws + off; off += (size_t)N * HEADS;
  float* denom  = ws + off; off += (size_t)N * HEADS;
  float* ealpha = ws + off; off += (size_t)EN * HEADS;
  (void)ws_size; (void)n_in; (void)out_size;

  // 1) projection GEMM on the WMMA pipe
  {
    const int tilesM = (N + 15) / 16;
    const int tiles  = tilesM * (HC / 16);
    const int blocks = (tiles + 7) / 8;          // 8 waves per 256-thr block
    gat_gemm_wmma<<<blocks, 256, 0, stream>>>(x, W, xh, N);
  }
  // 2) per-node attention dots
  gat_att_dots<<<(N * HEADS + 255) / 256, 256, 0, stream>>>(
      xh, att_src, att_dst, a_src, a_dst, N);
  // 3) init accumulators
  {
    const int out_n = N * OUT_C;
    const int nh    = N * HEADS;
    const int mx    = out_n > nh ? out_n : nh;
    gat_init<<<(mx + 255) / 256, 256, 0, stream>>>(out, amax, denom, out_n, nh);
  }
  // 4) alpha + leaky-relu + segment max
  gat_edge_alpha_max<<<(EN + 255) / 256, 256, 0, stream>>>(
      ei, a_src, a_dst, ealpha, amax, E, EN);
  // 5) exp + segment sum
  gat_edge_exp_sum<<<(EN + 255) / 256, 256, 0, stream>>>(
      ei, ealpha, amax, denom, E, EN);
  // 6) weighted scatter-add (head mean folded in)
  {
    const long long threads = (long long)EN * OUT_C;
    const int blocks = (int)((threads + 255) / 256);
    gat_scatter<<<blocks, 256, 0, stream>>>(ei, xh, ealpha, denom, out, E, EN);
  }
  // 7) bias + relu
  gat_finalize<<<(N * OUT_C + 255) / 256, 256, 0, stream>>>(
      out, bias_gat, bias, N * OUT_C);
}